// GATModel_29059748725071
// MI455X (gfx1250) — compile-verified
//
#include <hip/hip_runtime.h>
#include <hip/hip_bf16.h>
#include <math.h>

#define HDIM 4
#define DDIM 64
#define F 256          // in/out feature width for both layers (IN_DIM = H*D = 256)
#define NEG_SLOPE 0.2f

typedef __attribute__((ext_vector_type(2))) float v2f;
typedef __attribute__((ext_vector_type(8))) float v8f;

// ---------- order-preserving float <-> uint encoding for atomicMax ----------
__device__ __forceinline__ unsigned fenc(float f) {
    unsigned u = __float_as_uint(f);
    return (u & 0x80000000u) ? ~u : (u | 0x80000000u);
}
__device__ __forceinline__ float fdec(unsigned u) {
    return __uint_as_float((u & 0x80000000u) ? (u & 0x7FFFFFFFu) : ~u);
}
__device__ __forceinline__ float lrelu(float v) {
    return v > 0.f ? v : NEG_SLOPE * v;
}

// ---------- GEMM: h[M,256] = x[M,256] @ W[256,256], f32 WMMA 16x16x4 ----------
// One wave per 16x16 output tile. M must be a multiple of 16 (50000 = 3125*16).
__global__ void gat_gemm_wmma(const float* __restrict__ x,
                              const float* __restrict__ W,
                              float* __restrict__ h, int M) {
    const int wave = (int)((blockIdx.x * blockDim.x + threadIdx.x) >> 5);
    const int lane = (int)(threadIdx.x & 31);
    const int tilesN = F / 16;                 // 16
    const int tileM = wave / tilesN;
    const int tileN = wave % tilesN;
    if (tileM * 16 >= M) return;               // wave-uniform exit; EXEC stays all-1s

    const int mrow  = tileM * 16 + (lane & 15);      // A row this lane loads
    const int ncol  = tileN * 16 + (lane & 15);      // B/C column this lane owns
    const int khalf = (lane >> 4) << 1;              // 0 for lanes 0-15, 2 for 16-31

    v8f c = {};
    #pragma unroll 4
    for (int k0 = 0; k0 < F; k0 += 4) {
        const int kb = k0 + khalf;
        v2f a, b;
        // A 16x4 frag: lanes0-15 -> K={k0,k0+1}, lanes16-31 -> K={k0+2,k0+3}
        a.x = x[mrow * F + kb];
        a.y = x[mrow * F + kb + 1];
        // B 4x16 frag: VGPR v holds row K=kb+v, N striped across lane&15
        b.x = W[kb * F + ncol];
        b.y = W[(kb + 1) * F + ncol];
        c = __builtin_amdgcn_wmma_f32_16x16x4_f32(
                /*neg_a=*/false, a, /*neg_b=*/false, b,
                /*c_mod=*/(short)0, c, /*reuse_a=*/false, /*reuse_b=*/false);
    }
    // C/D layout: VGPR r, lanes0-15 -> M=r, lanes16-31 -> M=r+8, N=lane&15
    const int rbase = tileM * 16 + ((lane >> 4) << 3);
    #pragma unroll
    for (int r = 0; r < 8; ++r)
        h[(rbase + r) * F + (tileN * 16 + (lane & 15))] = c[r];
}

// ---------- el/er: per-(node,head) dot products with al/ar ----------
__global__ void gat_attn_coef(const float* __restrict__ h,
                              const float* __restrict__ al,
                              const float* __restrict__ ar,
                              float* __restrict__ el, float* __restrict__ er,
                              int Ncnt) {
    int idx = blockIdx.x * blockDim.x + threadIdx.x;   // n*H + head
    if (idx >= Ncnt * HDIM) return;
    int n = idx / HDIM, hd = idx % HDIM;
    const float* hp  = h  + (size_t)n * F + hd * DDIM;
    const float* alp = al + hd * DDIM;
    const float* arp = ar + hd * DDIM;
    float sl = 0.f, sr = 0.f;
    #pragma unroll 8
    for (int d = 0; d < DDIM; ++d) { float v = hp[d]; sl += v * alp[d]; sr += v * arp[d]; }
    el[idx] = sl; er[idx] = sr;
}

// ---------- init per-node softmax state ----------
__global__ void gat_softmax_init(unsigned* __restrict__ mx, float* __restrict__ sm, int cnt) {
    int idx = blockIdx.x * blockDim.x + threadIdx.x;
    if (idx < cnt) { mx[idx] = fenc(-INFINITY); sm[idx] = 0.f; }
}

// ---------- pass 1: segment max over incoming edges ----------
__global__ void gat_edge_max(const int* __restrict__ src, const int* __restrict__ dst,
                             const float* __restrict__ el, const float* __restrict__ er,
                             unsigned* __restrict__ mx, int E) {
    int idx = blockIdx.x * blockDim.x + threadIdx.x;   // e*H + head
    if (idx >= E * HDIM) return;
    int e = idx / HDIM, hd = idx % HDIM;
    int s = src[e], d = dst[e];
    float v = lrelu(el[s * HDIM + hd] + er[d * HDIM + hd]);
    atomicMax(&mx[d * HDIM + hd], fenc(v));
}

// ---------- pass 2: segment sum of exp(e - max) ----------
__global__ void gat_edge_sum(const int* __restrict__ src, const int* __restrict__ dst,
                             const float* __restrict__ el, const float* __restrict__ er,
                             const unsigned* __restrict__ mx, float* __restrict__ sm, int E) {
    int idx = blockIdx.x * blockDim.x + threadIdx.x;
    if (idx >= E * HDIM) return;
    int e = idx / HDIM, hd = idx % HDIM;
    int s = src[e], d = dst[e];
    float v = lrelu(el[s * HDIM + hd] + er[d * HDIM + hd]);
    float ee = __expf(v - fdec(mx[d * HDIM + hd]));
    atomicAdd(&sm[d * HDIM + hd], ee);
}

// ---------- init output with bias (broadcast b[256] over rows) ----------
__global__ void gat_bias_init(float* __restrict__ out, const float* __restrict__ b, int Ncnt) {
    int idx = blockIdx.x * blockDim.x + threadIdx.x;
    if (idx < Ncnt * F) out[idx] = b[idx & (F - 1)];
}

// ---------- pass 3: weighted aggregation; 64 lanes per edge, float4 per lane ----
__global__ void gat_aggregate(const int* __restrict__ src, const int* __restrict__ dst,
                              const float* __restrict__ h,
                              const float* __restrict__ el, const float* __restrict__ er,
                              const unsigned* __restrict__ mx, const float* __restrict__ sm,
                              float* __restrict__ out, int E) {
    const int group  = (int)((blockIdx.x * blockDim.x + threadIdx.x) >> 6);
    const int lane64 = (int)(threadIdx.x & 63);
    const int stride = (int)((gridDim.x * blockDim.x) >> 6);
    const int hd     = lane64 >> 4;                 // (lane64*4)/64
    for (int e = group; e < E; e += stride) {
        int s = src[e], d = dst[e];
        float v = lrelu(el[s * HDIM + hd] + er[d * HDIM + hd]);
        float alpha = __expf(v - fdec(mx[d * HDIM + hd])) / sm[d * HDIM + hd];
        const float4 hv = *(const float4*)(h + (size_t)s * F + lane64 * 4);
        float* op = out + (size_t)d * F + lane64 * 4;
        atomicAdd(op + 0, alpha * hv.x);
        atomicAdd(op + 1, alpha * hv.y);
        atomicAdd(op + 2, alpha * hv.z);
        atomicAdd(op + 3, alpha * hv.w);
    }
}

// ---------- elementwise ReLU ----------
__global__ void gat_relu(float* __restrict__ x, int cnt) {
    int idx = blockIdx.x * blockDim.x + threadIdx.x;
    if (idx < cnt) { float v = x[idx]; x[idx] = v > 0.f ? v : 0.f; }
}

extern "C" void kernel_launch(void* const* d_in, const int* in_sizes, int n_in,
                              void* d_out, int out_size, void* d_ws, size_t ws_size,
                              hipStream_t stream) {
    // inputs (setup_inputs order): features, src, dst, W1, al1, ar1, b1, W2, al2, ar2, b2
    const float* feat = (const float*)d_in[0];
    const int*   src  = (const int*)  d_in[1];
    const int*   dst  = (const int*)  d_in[2];
    const float* W1   = (const float*)d_in[3];
    const float* al1  = (const float*)d_in[4];
    const float* ar1  = (const float*)d_in[5];
    const float* b1   = (const float*)d_in[6];
    const float* W2   = (const float*)d_in[7];
    const float* al2  = (const float*)d_in[8];
    const float* ar2  = (const float*)d_in[9];
    const float* b2   = (const float*)d_in[10];
    float* outF = (float*)d_out;

    const int N = in_sizes[0] / F;      // 50000
    const int E = in_sizes[1];          // 850000

    // workspace carve-out
    size_t off = 0;
    auto carve = [&](size_t bytes) -> char* {
        char* p = (char*)d_ws + off;
        off += (bytes + 255) & ~(size_t)255;
        return p;
    };
    float*    hbuf = (float*)   carve((size_t)N * F * sizeof(float));  // 51.2 MB
    float*    xbuf = (float*)   carve((size_t)N * F * sizeof(float));  // 51.2 MB
    float*    el   = (float*)   carve((size_t)N * HDIM * sizeof(float));
    float*    er   = (float*)   carve((size_t)N * HDIM * sizeof(float));
    unsigned* mx   = (unsigned*)carve((size_t)N * HDIM * sizeof(unsigned));
    float*    sm   = (float*)   carve((size_t)N * HDIM * sizeof(float));
    (void)ws_size; (void)n_in; (void)out_size;

    const int waves   = (N / 16) * (F / 16);              // 50000 tiles
    const int gemmBlk = (waves + 7) / 8;                  // 8 waves / 256-thread block
    const int nhBlk   = (N * HDIM + 255) / 256;
    const int ehBlk   = (E * HDIM + 255) / 256;
    const int nfBlk   = (N * F + 255) / 256;
    const int aggBlk  = (E + 3) / 4;                      // 4 edge-groups per block

    auto layer = [&](const float* x, const float* W, const float* al_, const float* ar_,
                     const float* b_, float* out) {
        gat_gemm_wmma   <<<gemmBlk, 256, 0, stream>>>(x, W, hbuf, N);
        gat_attn_coef   <<<nhBlk,   256, 0, stream>>>(hbuf, al_, ar_, el, er, N);
        gat_softmax_init<<<nhBlk,   256, 0, stream>>>(mx, sm, N * HDIM);
        gat_edge_max    <<<ehBlk,   256, 0, stream>>>(src, dst, el, er, mx, E);
        gat_edge_sum    <<<ehBlk,   256, 0, stream>>>(src, dst, el, er, mx, sm, E);
        gat_bias_init   <<<nfBlk,   256, 0, stream>>>(out, b_, N);
        gat_aggregate   <<<aggBlk,  256, 0, stream>>>(src, dst, hbuf, el, er, mx, sm, out, E);
        gat_relu        <<<nfBlk,   256, 0, stream>>>(out, N * F);
    };

    layer(feat, W1, al1, ar1, b1, xbuf);   // layer 1 -> xbuf
    layer(xbuf, W2, al2, ar2, b2, outF);   // layer 2 -> d_out
}